// VisibleRebuildModule_57251914056190
// MI455X (gfx1250) — compile-verified
//
#include <hip/hip_runtime.h>
#include <hip/hip_bf16.h>

#define H_ 256
#define W_ 256
#define HW_ (H_*W_)
#define NK 68
#define OFS 6.0f

typedef __attribute__((ext_vector_type(2))) float v2f;
typedef __attribute__((ext_vector_type(4))) float f4;
typedef __attribute__((ext_vector_type(8))) float v8f;

__device__ __forceinline__ float wave_sum(float v) {
#pragma unroll
    for (int m = 16; m > 0; m >>= 1) v += __shfl_xor(v, m, 32);
    return v;
}

// ---------------------------------------------------------------------------
// Kernel 1: per-batch Kabsch. One wave32 per batch. Hm = AA^T * BB computed
// with V_WMMA_F32_16X16X4_F32 (17 chunks of K=4 keypoints) fed from a
// transposed, zero-padded LDS image (one ds_load_b64 per operand, no EXEC
// juggling), then a 3x3 one-sided Jacobi SVD; writes M = scale*R and t.
// ---------------------------------------------------------------------------
__global__ __launch_bounds__(32)
void kabsch_kernel(const float* __restrict__ Offset,
                   const float* __restrict__ Posmap,
                   const float* __restrict__ meanp,
                   const int*   __restrict__ uv,
                   float*       __restrict__ Mt)
{
    const int b    = blockIdx.x;
    const int lane = threadIdx.x;

    __shared__ float sSrc[NK][3];
    __shared__ float sDst[NK][3];
    // Transposed operand images for WMMA: [row/col r][k], rows 3..15 zero.
    // Row stride 72 floats (288 B) keeps &[rc][even k] 8-byte aligned.
    __shared__ __align__(16) float sAAr[16][72];
    __shared__ __align__(16) float sBBr[16][72];
    __shared__ float sHm[9];

    const float* offb = Offset + (size_t)b * 3 * HW_;
    const float* posb = Posmap + (size_t)b * 3 * HW_;

    // Zero the padded operand images (covers rows 3..15 and k padding)
    for (int i = lane; i < 16*72; i += 32) {
        (&sAAr[0][0])[i] = 0.f;
        (&sBBr[0][0])[i] = 0.f;
    }

    // Gather 68 keypoints
    for (int i = lane; i < NK; i += 32) {
        const int pix = uv[2*i] * W_ + uv[2*i + 1];
#pragma unroll
        for (int c = 0; c < 3; ++c) {
            sSrc[i][c] = fmaf(offb[c*HW_ + pix], OFS, meanp[c*HW_ + pix]);
            sDst[i][c] = posb[c*HW_ + pix];
        }
    }
    __syncthreads();

    // scale = sum||dst_i - dst_0|| / sum||src_i - src_0||
    float s1p = 0.f, s2p = 0.f;
    const float sx0 = sSrc[0][0], sy0 = sSrc[0][1], sz0 = sSrc[0][2];
    const float dx0 = sDst[0][0], dy0 = sDst[0][1], dz0 = sDst[0][2];
    for (int i = lane; i < NK; i += 32) {
        float ax = sSrc[i][0]-sx0, ay = sSrc[i][1]-sy0, az = sSrc[i][2]-sz0;
        s1p += sqrtf(ax*ax + ay*ay + az*az);
        float bx = sDst[i][0]-dx0, by = sDst[i][1]-dy0, bz = sDst[i][2]-dz0;
        s2p += sqrtf(bx*bx + by*by + bz*bz);
    }
    const float scale = wave_sum(s2p) / wave_sum(s1p);

    // centroids of A = scale*src and of dst
    float sa[3] = {0,0,0}, sb[3] = {0,0,0};
    for (int i = lane; i < NK; i += 32) {
#pragma unroll
        for (int c = 0; c < 3; ++c) {
            sa[c] += scale * sSrc[i][c];
            sb[c] += sDst[i][c];
        }
    }
    float ma[3], mb[3];
#pragma unroll
    for (int c = 0; c < 3; ++c) {
        ma[c] = wave_sum(sa[c]) * (1.0f/NK);
        mb[c] = wave_sum(sb[c]) * (1.0f/NK);
    }

    // Centered operands, written transposed: sAAr[c][i], sBBr[c][i]
    for (int i = lane; i < NK; i += 32) {
#pragma unroll
        for (int c = 0; c < 3; ++c) {
            sAAr[c][i] = scale * sSrc[i][c] - ma[c];
            sBBr[c][i] = sDst[i][c] - mb[c];
        }
    }
    __syncthreads();

    // Hm(3x3, padded to 16x16) = sum_k AA_k (outer) BB_k via WMMA f32 16x16x4.
    // A 16x4 layout: lanes 0-15 row M=lane hold K0(v.x),K1(v.y); lanes 16-31 K2,K3.
    // B 4x16 layout: lanes 0-15 col N=lane hold K0,K1; lanes 16-31 K2,K3.
    const int rc = lane & 15;          // row of A / col of B
    const int kh = (lane >> 4) * 2;    // K pair selected by lane half
    v8f acc = {};
#pragma unroll
    for (int ch = 0; ch < 17; ++ch) {  // 17 * 4 = 68 keypoints
        const int k0 = ch*4 + kh;      // always even -> 8B-aligned LDS reads
        const v2f av = *(const v2f*)&sAAr[rc][k0];
        const v2f bv = *(const v2f*)&sBBr[rc][k0];
        acc = __builtin_amdgcn_wmma_f32_16x16x4_f32(false, av, false, bv,
                                                    (short)0, acc, false, false);
    }
    // D layout: VGPR r, lane l<16 -> D[M=r][N=l]. Extract 3x3 block.
    if (lane < 3) {
        sHm[0*3 + lane] = acc[0];
        sHm[1*3 + lane] = acc[1];
        sHm[2*3 + lane] = acc[2];
    }
    __syncthreads();

    // 3x3 SVD via one-sided Jacobi (uniform across wave): Hm = U * diag(sig) * V^T
    float G[3][3], V[3][3];
#pragma unroll
    for (int i = 0; i < 3; ++i)
#pragma unroll
        for (int j = 0; j < 3; ++j) {
            G[i][j] = sHm[i*3 + j];
            V[i][j] = (i == j) ? 1.f : 0.f;
        }

    for (int sweep = 0; sweep < 10; ++sweep) {
#pragma unroll
        for (int pair = 0; pair < 3; ++pair) {
            const int p = (pair == 2) ? 1 : 0;
            const int q = (pair == 0) ? 1 : 2;
            float alpha = 0.f, beta = 0.f, gamma = 0.f;
#pragma unroll
            for (int i = 0; i < 3; ++i) {
                alpha += G[i][p]*G[i][p];
                beta  += G[i][q]*G[i][q];
                gamma += G[i][p]*G[i][q];
            }
            if (fabsf(gamma) > 1e-20f) {
                const float zeta = (beta - alpha) / (2.f * gamma);
                const float tt = copysignf(1.f, zeta) /
                                 (fabsf(zeta) + sqrtf(1.f + zeta*zeta));
                const float cc = 1.f / sqrtf(1.f + tt*tt);
                const float ss = cc * tt;
#pragma unroll
                for (int i = 0; i < 3; ++i) {
                    float gp = G[i][p], gq = G[i][q];
                    G[i][p] = cc*gp - ss*gq;  G[i][q] = ss*gp + cc*gq;
                    float vp = V[i][p], vq = V[i][q];
                    V[i][p] = cc*vp - ss*vq;  V[i][q] = ss*vp + cc*vq;
                }
            }
        }
    }

    float inv_sig[3];
#pragma unroll
    for (int k = 0; k < 3; ++k) {
        float ss2 = G[0][k]*G[0][k] + G[1][k]*G[1][k] + G[2][k]*G[2][k];
        inv_sig[k] = (ss2 > 1e-30f) ? (1.f / sqrtf(ss2)) : 0.f;
    }

    // R = V * U^T  (U[:,k] = G[:,k]/sig_k);  M = scale * R
    float M[3][3];
#pragma unroll
    for (int e = 0; e < 3; ++e)
#pragma unroll
        for (int d = 0; d < 3; ++d) {
            float r = 0.f;
#pragma unroll
            for (int k = 0; k < 3; ++k) r += V[e][k] * G[d][k] * inv_sig[k];
            M[e][d] = scale * r;
        }

    // t = mean_i( dst_i - M * src_i )
    float tp[3] = {0,0,0};
    for (int i = lane; i < NK; i += 32) {
        const float px = sSrc[i][0], py = sSrc[i][1], pz = sSrc[i][2];
#pragma unroll
        for (int e = 0; e < 3; ++e)
            tp[e] += sDst[i][e] - (M[e][0]*px + M[e][1]*py + M[e][2]*pz);
    }
    float tv[3];
#pragma unroll
    for (int e = 0; e < 3; ++e) tv[e] = wave_sum(tp[e]) * (1.0f/NK);

    if (lane == 0) {
        float* o = Mt + b*12;
#pragma unroll
        for (int e = 0; e < 3; ++e)
#pragma unroll
            for (int d = 0; d < 3; ++d) o[e*3 + d] = M[e][d];
        o[9] = tv[0]; o[10] = tv[1]; o[11] = tv[2];
    }
}

// ---------------------------------------------------------------------------
// Kernel 2: bandwidth-bound transform (the ~8.6 us roofline part).
// 4 pixels/thread, B128 per plane. b derives from blockIdx only -> M,t
// loads are scalar. Non-temporal on the 200 MB stream; mean_posmap stays
// L2-resident (768 KB in 192 MB L2, reused by all 128 batches).
// ---------------------------------------------------------------------------
__global__ __launch_bounds__(256)
void transform_kernel(const f4* __restrict__ Offset,
                      const f4* __restrict__ meanp,
                      const float* __restrict__ Mt,
                      f4* __restrict__ out)
{
    const int b   = blockIdx.x >> 6;                          // 64 blocks / batch
    const int pix = ((blockIdx.x & 63) << 8) + threadIdx.x;   // 0..16383 (f4 units)
    const int PL  = HW_ / 4;                                  // plane stride in f4
    const size_t base = (size_t)b * 3 * PL;

    const float* mt = Mt + b * 12;
    const float m00 = mt[0], m01 = mt[1], m02 = mt[2];
    const float m10 = mt[3], m11 = mt[4], m12 = mt[5];
    const float m20 = mt[6], m21 = mt[7], m22 = mt[8];
    const float t0  = mt[9], t1  = mt[10], t2 = mt[11];

    f4 p0 = __builtin_nontemporal_load(&Offset[base          + pix]);
    f4 p1 = __builtin_nontemporal_load(&Offset[base +   PL   + pix]);
    f4 p2 = __builtin_nontemporal_load(&Offset[base + 2*PL   + pix]);
    const f4 c0 = meanp[pix];
    const f4 c1 = meanp[PL + pix];
    const f4 c2 = meanp[2*PL + pix];

    const f4 q0 = p0 * OFS + c0;
    const f4 q1 = p1 * OFS + c1;
    const f4 q2 = p2 * OFS + c2;

    f4 o0 = q0 * m00 + q1 * m01 + q2 * m02 + t0;
    f4 o1 = q0 * m10 + q1 * m11 + q2 * m12 + t1;
    f4 o2 = q0 * m20 + q1 * m21 + q2 * m22 + t2;

    __builtin_nontemporal_store(o0, &out[base          + pix]);
    __builtin_nontemporal_store(o1, &out[base +   PL   + pix]);
    __builtin_nontemporal_store(o2, &out[base + 2*PL   + pix]);
}

extern "C" void kernel_launch(void* const* d_in, const int* in_sizes, int n_in,
                              void* d_out, int out_size, void* d_ws, size_t ws_size,
                              hipStream_t stream) {
    const float* Offset = (const float*)d_in[0];   // (B,3,256,256)
    const float* Posmap = (const float*)d_in[1];   // (B,3,256,256)
    const float* meanp  = (const float*)d_in[2];   // (3,256,256)
    const int*   uv     = (const int*)  d_in[3];   // (68,2) int32
    float* out = (float*)d_out;                    // (B,3,256,256)
    float* Mt  = (float*)d_ws;                     // B * 12 floats

    const int B = in_sizes[0] / (3 * HW_);

    kabsch_kernel<<<B, 32, 0, stream>>>(Offset, Posmap, meanp, uv, Mt);
    transform_kernel<<<B * 64, 256, 0, stream>>>((const f4*)Offset,
                                                 (const f4*)meanp,
                                                 Mt, (f4*)out);
}